// CRFDecoder_1511828488941
// MI455X (gfx1250) — compile-verified
//
#include <hip/hip_runtime.h>

// CRF forward (alpha) recursion on gfx1250 via exp-domain WMMA:
//   new_alpha = m + log( exp(alpha - m) @ exp(trans) ) + x_t
// exp(trans) computed once; per step the O(B*L*L) reduction runs on chained
// V_WMMA_F32_16X16X4_F32 (fp32), leaving only O(B*L) exp/log on the VALU.
// exp/log use raw v_exp_f32/v_log_f32 (operands provably out of denorm range).

typedef __attribute__((ext_vector_type(2))) float v2f;
typedef __attribute__((ext_vector_type(8))) float v8f;

#define NEGV (-10000.0f)
#define LOG2E 1.44269504088896f
#define LN2   0.693147180559945f

__device__ __forceinline__ float fast_exp(float x) {
  return __builtin_amdgcn_exp2f(x * LOG2E);   // v_exp_f32
}
__device__ __forceinline__ float fast_log(float x) {
  return __builtin_amdgcn_logf(x) * LN2;      // v_log_f32
}

__global__ __launch_bounds__(128) void crf_forward_wmma(
    const float* __restrict__ X,      // [256, 512, 64]
    const float* __restrict__ trans,  // [64, 64]
    float* __restrict__ out)          // [256, 64]
{
  constexpr int T = 512, L = 64;
  constexpr int AS = 68;  // padded row stride (floats) for alpha / S tiles

  __shared__ float texp[64 * 64];              // exp(transition), 16 KB
  __shared__ float alpha[16 * AS];             // current alpha, batch tile of 16
  __shared__ float sbuf[16 * AS];              // S = exp(alpha - m)
  __shared__ __align__(16) float mfull[16];    // row max of alpha
  __shared__ __align__(16) float rsum[16];     // row sum of S (tag-B column fix)

  const int tid  = threadIdx.x;         // 0..127 (4 waves)
  const int wave = tid >> 5;
  const int lane = tid & 31;
  const int half = lane >> 4;           // wave32 half
  const int n    = lane & 15;
  const int colBase = wave * 16;        // this wave's nxt-tag chunk
  const int rrow = tid >> 3;            // reduction-phase row (8 lanes/row,
  const int jj   = tid & 7;             //  contiguous + 8-aligned in a wave)
  const int c0   = jj * 8;
  const long baseB = (long)blockIdx.x * 16;
  const bool isColB = (colBase + n) == 0;  // lane-invariant across r-loop

  // ---- one-time: Texp = exp(trans) into LDS; init alpha tile ----
  for (int i = tid; i < 64 * 64; i += 128)
    texp[i] = fast_exp(trans[i]);      // exp(-10000) -> 0 exactly
  for (int i = tid; i < 16 * 64; i += 128) {
    int r = i >> 6, c = i & 63;
    alpha[r * AS + c] = (c == 0) ? 0.0f : NEGV;  // init_alpha: B tag = 0
  }
  __syncthreads();

  // ---- B-matrix fragments (4x16 F32 per WMMA), resident all T steps ----
  // VGPR0: lanes0-15 K=k, lanes16-31 K=k+2; VGPR1: K=k+1 / K=k+3.
  v2f bf[16];
#pragma unroll
  for (int ch = 0; ch < 16; ++ch) {
    const int k = 4 * ch + 2 * half;
    v2f b;
    b.x = texp[(k + 0) * 64 + colBase + n];
    b.y = texp[(k + 1) * 64 + colBase + n];
    bf[ch] = b;
  }

  for (int t = 0; t < T; ++t) {
    __syncthreads();  // alpha(t-1) writes visible; sbuf(t-1) reads done

    // ---- row max + S = exp(alpha - m) + row sum, registers + shuffles only
    float av[8];
#pragma unroll
    for (int i = 0; i < 8; ++i) av[i] = alpha[rrow * AS + c0 + i];
    float mx = av[0];
#pragma unroll
    for (int i = 1; i < 8; ++i) mx = fmaxf(mx, av[i]);
    mx = fmaxf(mx, __shfl_xor(mx, 1, 32));   // 8-lane group reduce (same wave)
    mx = fmaxf(mx, __shfl_xor(mx, 2, 32));
    mx = fmaxf(mx, __shfl_xor(mx, 4, 32));
    if (jj == 0) mfull[rrow] = mx;
    float sm = 0.0f;
#pragma unroll
    for (int i = 0; i < 8; ++i) {
      float s = fast_exp(av[i] - mx);
      sbuf[rrow * AS + c0 + i] = s;
      sm += s;
    }
    sm += __shfl_xor(sm, 1, 32);
    sm += __shfl_xor(sm, 2, 32);
    sm += __shfl_xor(sm, 4, 32);
    if (jj == 0) rsum[rrow] = sm;
    __syncthreads();

    // emission loads for this wave's C tile (issued early to hide latency)
    float xv[8];
#pragma unroll
    for (int r = 0; r < 8; ++r) {
      const int M = r + half * 8;
      xv[r] = X[((baseB + M) * T + t) * (long)L + colBase + n];
    }
    if (t + 4 < T)
      __builtin_prefetch(&X[((baseB + half * 8) * T + (t + 4)) * (long)L + colBase + n], 0, 3);

    // ---- P = S @ Texp via 16 chained v_wmma_f32_16x16x4_f32 (K=64)
    // A 16x4 F32: lane n holds M=n; lanes<16: K=4c,4c+1; lanes>=16: K=4c+2,4c+3
    v8f acc = {};
#pragma unroll
    for (int ch = 0; ch < 16; ++ch) {
      const int k = 4 * ch + 2 * half;          // even -> 8B aligned LDS read
      v2f a = *(const v2f*)&sbuf[n * AS + k];
      acc = __builtin_amdgcn_wmma_f32_16x16x4_f32(
          /*neg_a=*/false, a, /*neg_b=*/false, bf[ch],
          /*c_mod=*/(short)0, acc, /*reuse_a=*/false, /*reuse_b=*/false);
    }

    // row maxes for this lane's 8 output rows: two vector LDS reads
    const float4 ma = *(const float4*)&mfull[half * 8];
    const float4 mb = *(const float4*)&mfull[half * 8 + 4];
    const float mr[8] = {ma.x, ma.y, ma.z, ma.w, mb.x, mb.y, mb.z, mb.w};

    // ---- new_alpha = m + log(P) + x_t ; branch hoisted out of the r-loop
    if (isColB) {
      // trans[:,B] == -10000 uniformly: logsumexp factors exactly
      const float4 ra = *(const float4*)&rsum[half * 8];
      const float4 rb = *(const float4*)&rsum[half * 8 + 4];
      const float rs[8] = {ra.x, ra.y, ra.z, ra.w, rb.x, rb.y, rb.z, rb.w};
#pragma unroll
      for (int r = 0; r < 8; ++r) {
        const int M = r + half * 8;
        alpha[M * AS + colBase + n] = mr[r] - 10000.0f + fast_log(rs[r]) + xv[r];
      }
    } else {
#pragma unroll
      for (int r = 0; r < 8; ++r) {
        const int M = r + half * 8;             // C layout: VGPR r -> M=r / r+8
        alpha[M * AS + colBase + n] = mr[r] + fast_log(acc[r]) + xv[r];
      }
    }
  }

  __syncthreads();
  for (int i = tid; i < 16 * 64; i += 128) {
    int r = i >> 6, c = i & 63;
    out[(baseB + r) * 64 + c] = alpha[r * AS + c];
  }
}

extern "C" void kernel_launch(void* const* d_in, const int* in_sizes, int n_in,
                              void* d_out, int out_size, void* d_ws, size_t ws_size,
                              hipStream_t stream) {
  (void)in_sizes; (void)n_in; (void)out_size; (void)d_ws; (void)ws_size;
  const float* X     = (const float*)d_in[0];  // [256,512,64] fp32
  const float* trans = (const float*)d_in[1];  // [64,64] fp32
  float* out = (float*)d_out;                  // [256,64] fp32
  // One workgroup per 16-batch tile; 4 waves, each owning a 16-tag column chunk.
  crf_forward_wmma<<<dim3(256 / 16), dim3(128), 0, stream>>>(X, trans, out);
}